// NARGNN_54795192762572
// MI455X (gfx1250) — compile-verified
//
#include <hip/hip_runtime.h>
#include <hip/hip_bf16.h>

typedef __attribute__((ext_vector_type(2))) float v2f;
typedef __attribute__((ext_vector_type(8))) float v8f;

#define HDIM 128
#define NBR_CAP 64
#define OVF_CAP 8192

// ---------------------------------------------------------------------------
// Pack a 128x128 weight matrix into per-lane WMMA B-fragment order:
// record (step, nt) = 32 lanes x 2 dwords:  out[((step*8+nt)*32 + L)*2 + {0,1}]
//   = W[step*4 + 2*(L>>4) + {0,1}][nt*16 + (L&15)]
// One ds_load_b64 per fragment, conflict-free banks (lane -> bank pair 2L,2L+1).
// ---------------------------------------------------------------------------
__global__ __launch_bounds__(256) void gin_pack_w(
    const float* __restrict__ W, float* __restrict__ out)
{
    const int P = blockIdx.x * blockDim.x + threadIdx.x;   // pair index 0..8191
    if (P >= 8192) return;
    const int step = P >> 8;
    const int nt   = (P >> 5) & 7;
    const int L    = P & 31;
    const int ka   = step * 4 + (L >> 4) * 2;
    const int n    = nt * 16 + (L & 15);
    out[2 * P]     = W[ka * HDIM + n];
    out[2 * P + 1] = W[(ka + 1) * HDIM + n];
}

// ---------------------------------------------------------------------------
// out[N x 128] = A[N x 128] @ W + bias (optional ReLU), W pre-packed.
// One wave per 16-row slab, 8 f32 accumulator tiles, fp32 WMMA 16x16x4.
// Weights staged to LDS with GLOBAL_LOAD_ASYNC_TO_LDS_B128 (ASYNCcnt path).
// ---------------------------------------------------------------------------
__global__ __launch_bounds__(256) void gin_gemm128_wmma(
    const float* __restrict__ A, const float* __restrict__ Wp,
    const float* __restrict__ bias, float* __restrict__ out,
    int nrows, int do_relu)
{
    __shared__ float sW[HDIM * HDIM];          // 64 KB packed weights

    const int tid = threadIdx.x;
    // async copy: memory -> LDS directly (no VGPR round-trip), 16B per lane
    for (int i = tid; i < (HDIM * HDIM) / 4; i += 256) {
        const unsigned ldsDst = (unsigned)(uintptr_t)(sW + 4 * (size_t)i);
        const float*   gSrc   = Wp + 4 * (size_t)i;
        asm volatile("global_load_async_to_lds_b128 %0, %1, off"
                     :: "v"(ldsDst), "v"(gSrc) : "memory");
    }
    asm volatile("s_wait_asynccnt 0" ::: "memory");
    __syncthreads();

    const int lane  = tid & 31;
    const int m16   = lane & 15;
    const int khalf = lane >> 4;
    const int rowTile = blockIdx.x * 8 + (tid >> 5);
    if (rowTile * 16 >= nrows) return;         // wave-uniform exit

    int arow = rowTile * 16 + m16;
    if (arow >= nrows) arow = nrows - 1;
    const float* Arow = A + (size_t)arow * HDIM + khalf * 2;
    const float* sB   = sW + lane * 2;         // 8-byte aligned lane base

    v8f acc[8] = {};
    for (int step = 0; step < 32; ++step) {
        // A fragment: lane holds A[m][step*4 + 2*khalf .. +1]
        const v2f a = *(const v2f*)(Arow + step * 4);
        const float* sBs = sB + step * 512;    // (step*8)*64 dwords
        #pragma unroll
        for (int nt = 0; nt < 8; ++nt) {
            const v2f b = *(const v2f*)(sBs + nt * 64);   // single ds_load_b64
            acc[nt] = __builtin_amdgcn_wmma_f32_16x16x4_f32(
                false, a, false, b, (short)0, acc[nt], false, false);
        }
    }

    const int outRowBase = rowTile * 16;
    #pragma unroll
    for (int nt = 0; nt < 8; ++nt) {
        const int nc = nt * 16 + m16;
        const float bv = bias[nc];
        #pragma unroll
        for (int v = 0; v < 8; ++v) {
            const int mr = outRowBase + v + khalf * 8;    // C/D layout
            if (mr < nrows) {
                float val = acc[nt][v] + bv;
                if (do_relu) val = fmaxf(val, 0.0f);
                out[(size_t)mr * HDIM + nc] = val;
            }
        }
    }
}

// --------------------------- neighbor-list build ---------------------------
__global__ __launch_bounds__(256) void gin_init_deg(int* __restrict__ deg,
                                                    int* __restrict__ novf, int n)
{
    const int i = blockIdx.x * blockDim.x + threadIdx.x;
    if (i < n) deg[i] = 0;
    if (i == 0) *novf = 0;
}

__global__ __launch_bounds__(256) void gin_build_buckets(
    const int* __restrict__ src, const int* __restrict__ dst,
    int* __restrict__ deg, int* __restrict__ bkt,
    int* __restrict__ ovf, int* __restrict__ novf, int nedges)
{
    const int e = blockIdx.x * blockDim.x + threadIdx.x;
    if (e >= nedges) return;
    const int s = src[e];
    const int d = dst[e];
    const int slot = atomicAdd(&deg[d], 1);
    if (slot < NBR_CAP) {
        bkt[(size_t)d * NBR_CAP + slot] = s;
    } else {
        const int o = atomicAdd(novf, 1);
        if (o < OVF_CAP) { ovf[2 * o] = s; ovf[2 * o + 1] = d; }
    }
}

// ---------------------------------------------------------------------------
// z[i] = h[i] + sum_{j in N(i)} h[j]   (wave per node, atomic-free, L2-resident)
// ---------------------------------------------------------------------------
__global__ __launch_bounds__(256) void gin_gather_sum(
    const float* __restrict__ h, const int* __restrict__ deg,
    const int* __restrict__ bkt, float* __restrict__ z, int n)
{
    const int node = blockIdx.x * 8 + (threadIdx.x >> 5);
    const int lane = threadIdx.x & 31;
    if (node >= n) return;
    int d = deg[node];
    if (d > NBR_CAP) d = NBR_CAP;
    float4 acc = ((const float4*)(h + (size_t)node * HDIM))[lane];
    const int* b = bkt + (size_t)node * NBR_CAP;
    for (int j = 0; j < d; ++j) {
        const int s = b[j];
        const float4 v = ((const float4*)(h + (size_t)s * HDIM))[lane];
        acc.x += v.x; acc.y += v.y; acc.z += v.z; acc.w += v.w;
    }
    ((float4*)(z + (size_t)node * HDIM))[lane] = acc;
}

// Overflow fixup (normally zero iterations): z[d] += h[s] with f32 atomics.
__global__ __launch_bounds__(256) void gin_overflow_fix(
    const float* __restrict__ h, const int* __restrict__ ovf,
    const int* __restrict__ novf, float* __restrict__ z)
{
    int nv = *novf;
    if (nv > OVF_CAP) nv = OVF_CAP;
    const int total = nv * 32;
    for (int i = blockIdx.x * blockDim.x + threadIdx.x; i < total;
         i += gridDim.x * blockDim.x) {
        const int e = i >> 5, lane = i & 31;
        const int s = ovf[2 * e], d = ovf[2 * e + 1];
        const float4 v = ((const float4*)(h + (size_t)s * HDIM))[lane];
        float* zp = z + (size_t)d * HDIM + lane * 4;
        atomicAdd(zp + 0, v.x); atomicAdd(zp + 1, v.y);
        atomicAdd(zp + 2, v.z); atomicAdd(zp + 3, v.w);
    }
}

// ---------------------------------------------------------------------------
// logits[i] = h[i] . ro_W + ro_b   (wave per node, wave32 shfl reduction)
// ---------------------------------------------------------------------------
__global__ __launch_bounds__(256) void gin_readout(
    const float* __restrict__ h, const float* __restrict__ roW,
    const float* __restrict__ rob, float* __restrict__ logits, int n)
{
    const int node = blockIdx.x * 8 + (threadIdx.x >> 5);
    const int lane = threadIdx.x & 31;
    if (node >= n) return;
    const float4 v = ((const float4*)(h + (size_t)node * HDIM))[lane];
    const float4 w = ((const float4*)roW)[lane];
    float p = v.x * w.x + v.y * w.y + v.z * w.z + v.w * w.w;
    #pragma unroll
    for (int off = 16; off > 0; off >>= 1) p += __shfl_down(p, off, 32);
    if (lane == 0) logits[node] = p + rob[0];
}

extern "C" void kernel_launch(void* const* d_in, const int* in_sizes, int n_in,
                              void* d_out, int out_size, void* d_ws, size_t ws_size,
                              hipStream_t stream) {
    const float* x    = (const float*)d_in[0];
    const int*   eidx = (const int*)d_in[1];
    const float* embW = (const float*)d_in[3];
    const float* embB = (const float*)d_in[4];
    const float* W1   = (const float*)d_in[5];
    const float* b1   = (const float*)d_in[6];
    const float* W2   = (const float*)d_in[7];
    const float* b2   = (const float*)d_in[8];
    const float* roW  = (const float*)d_in[9];
    const float* rob  = (const float*)d_in[10];
    float* out = (float*)d_out;

    const int H = HDIM;
    const int N = in_sizes[2];
    const int E = in_sizes[1] / 2;
    const int L = in_sizes[5] / (H * H);
    const int* src = eidx;
    const int* dst = eidx + E;

    // workspace layout
    float* h    = (float*)d_ws;                    // N*128
    float* z    = h + (size_t)N * H;               // N*128
    float* pW   = z + (size_t)N * H;               // (1+2L)*16384 packed weights
    int*   deg  = (int*)(pW + (size_t)(1 + 2 * L) * H * H);  // N
    int*   bkt  = deg + N;                         // N*NBR_CAP
    int*   ovf  = bkt + (size_t)N * NBR_CAP;       // 2*OVF_CAP
    int*   novf = ovf + 2 * OVF_CAP;               // 1

    dim3 blk256(256);
    dim3 packGrid(8192 / 256);
    const int rowTiles = (N + 15) / 16;
    dim3 gemmGrid((rowTiles + 7) / 8);
    dim3 nodeGrid((N + 7) / 8);
    dim3 edgeGrid((E + 255) / 256);
    dim3 initGrid((N + 255) / 256);
    dim3 fixGrid(32);

    // pack all weight matrices into WMMA fragment order (slot 0=emb, 1+l=W1, 1+L+l=W2)
    gin_pack_w<<<packGrid, blk256, 0, stream>>>(embW, pW);
    for (int l = 0; l < L; ++l) {
        gin_pack_w<<<packGrid, blk256, 0, stream>>>(W1 + (size_t)l * H * H,
                                                    pW + (size_t)(1 + l) * H * H);
        gin_pack_w<<<packGrid, blk256, 0, stream>>>(W2 + (size_t)l * H * H,
                                                    pW + (size_t)(1 + L + l) * H * H);
    }

    // build per-destination neighbor lists (once; reused all layers)
    gin_init_deg<<<initGrid, blk256, 0, stream>>>(deg, novf, N);
    gin_build_buckets<<<edgeGrid, blk256, 0, stream>>>(src, dst, deg, bkt, ovf, novf, E);

    // h = x @ embed_W + embed_b
    gin_gemm128_wmma<<<gemmGrid, blk256, 0, stream>>>(x, pW, embB, h, N, 0);

    for (int l = 0; l < L; ++l) {
        // z = h + segment_sum(h[src], dst)
        gin_gather_sum<<<nodeGrid, blk256, 0, stream>>>(h, deg, bkt, z, N);
        gin_overflow_fix<<<fixGrid, blk256, 0, stream>>>(h, ovf, novf, z);
        // z = relu(z @ W1[l] + b1[l])   (in-place: each wave owns its 16 rows)
        gin_gemm128_wmma<<<gemmGrid, blk256, 0, stream>>>(
            z, pW + (size_t)(1 + l) * H * H, b1 + (size_t)l * H, z, N, 1);
        // h = z @ W2[l] + b2[l]; last layer writes straight into d_out[N..]
        float* hout = (l == L - 1) ? (out + N) : h;
        gin_gemm128_wmma<<<gemmGrid, blk256, 0, stream>>>(
            z, pW + (size_t)(1 + L + l) * H * H, b2 + (size_t)l * H, hout, N, 0);
    }

    gin_readout<<<nodeGrid, blk256, 0, stream>>>(out + N, roW, rob, out, N);
}